// SimplifiedRenderer_28501402977042
// MI455X (gfx1250) — compile-verified
//
#include <hip/hip_runtime.h>

typedef __attribute__((ext_vector_type(2))) float v2f;
typedef __attribute__((ext_vector_type(8))) float v8f;

#define N_G   512
#define IMG_W 256
#define IMG_H 256
#define F_FX  300.0f
#define F_CX  128.0f
#define F_CY  128.0f

// ---------------------------------------------------------------------------
// Kernel 1: project, stable depth-sort (rank by count), emit sorted AoS table.
// Per-gaussian record (12 floats, 48B, three 16B-aligned float4s):
//   [x0, x1, y0, y1,  neg_inv_2sig2, opacity, u, v,  r, g, b, pad]
// ---------------------------------------------------------------------------
__global__ void splat_prep(const float* __restrict__ pos,
                           const float* __restrict__ scl,
                           const float* __restrict__ col,
                           const float* __restrict__ opa,
                           const float* __restrict__ vm,
                           float* __restrict__ ws)
{
    __shared__ float sdepth[N_G];
    int i = threadIdx.x;

    float px = pos[3 * i + 0], py = pos[3 * i + 1], pz = pos[3 * i + 2];
    float cxr = vm[0] * px + vm[1] * py + vm[2]  * pz + vm[3];
    float cyr = vm[4] * px + vm[5] * py + vm[6]  * pz + vm[7];
    float czr = vm[8] * px + vm[9] * py + vm[10] * pz + vm[11];

    float zz = czr + 1e-8f;
    float sg = (zz > 0.0f) ? 1.0f : ((zz < 0.0f) ? -1.0f : 0.0f);
    float zc = fmaxf(fabsf(czr), 0.01f) * sg;
    float d  = -zc;                       // depth
    float u  = F_FX * cxr / d + F_CX;     // FX*x/-z + CX
    float v  = F_FX * (-cyr) / d + F_CY;  // FY*-y/-z + CY

    sdepth[i] = d;
    __syncthreads();

    // stable rank for argsort(-depth): larger depth first, index breaks ties
    int rank = 0;
    for (int j = 0; j < N_G; ++j) {
        float dj = sdepth[j];
        rank += (dj > d) || ((dj == d) && (j < i));
    }

    float sm     = (scl[3 * i] + scl[3 * i + 1] + scl[3 * i + 2]) * (1.0f / 3.0f);
    float radius = fminf(fmaxf(floorf(sm * F_FX / d), 1.0f), 20.0f);
    float xi = truncf(u), yi = truncf(v);
    float x0 = fmaxf(0.0f,          xi - radius);
    float x1 = fminf((float)IMG_W,  xi + radius + 1.0f);
    float y0 = fmaxf(0.0f,          yi - radius);
    float y1 = fminf((float)IMG_H,  yi + radius + 1.0f);
    float sig = fmaxf(radius * 0.5f, 1.0f);
    float n2s = -1.0f / (2.0f * sig * sig);
    float op  = opa[i];

    if (!(d > 0.0f)) {  // behind camera: empty box, zero opacity
        x0 = 1.0f; x1 = 0.0f; y0 = 1.0f; y1 = 0.0f;
        op = 0.0f; n2s = 0.0f; u = 0.0f; v = 0.0f;
    }

    float* g = ws + rank * 12;
    g[0] = x0;  g[1] = x1;  g[2] = y0;  g[3] = y1;
    g[4] = n2s; g[5] = op;  g[6] = u;   g[7] = v;
    g[8] = col[3 * i + 0]; g[9] = col[3 * i + 1]; g[10] = col[3 * i + 2];
    g[11] = 0.0f;
}

// ---------------------------------------------------------------------------
// Kernel 2: per-pixel back-to-front composite.
// 256 threads = 8 waves; each wave owns one 4x4 pixel tile (N = 16 pixels).
// Gaussians processed in 32 sorted chunks of 16 (M = 16) — dist^2 for the
// whole 16x16 (gaussian x pixel) tile comes from ONE v_wmma_f32_16x16x4_f32:
//   A[m] = [1, u'^2+v'^2, -2u', -2v'] ,  B[:,n] = [x'^2+y'^2, 1, x', y']
// (coords recentred on the tile to kill cancellation error).
// Lane n (<16) holds pixel n / gaussians 0..7; lane n+16 holds 8..15.
// Each half folds its run into affine (C,T); halves merge via shfl_xor(16).
// Sorted table staged to LDS with GLOBAL_LOAD_ASYNC_TO_LDS_B128 (ASYNCcnt).
// ---------------------------------------------------------------------------
__global__ void __launch_bounds__(256)
splat_render(const float* __restrict__ ws,
             const float* __restrict__ bg,
             float* __restrict__ out)
{
    __shared__ float4 sgs[N_G * 3];  // 24 KB sorted table

    __builtin_prefetch(ws, 0, 0);    // global_prefetch_b8

    const float4* wsv = (const float4*)ws;
    // gfx1250 async global->LDS path: no VGPR round-trip, ASYNCcnt-tracked.
    // GV mode: vdst = LDS byte address (low 32 bits of generic ptr), vaddr =
    // 64-bit global address, saddr = off.
    for (int k = threadIdx.x; k < N_G * 3; k += 256) {
        unsigned lds_off = (unsigned)(unsigned long long)&sgs[k];
        const void* gaddr = (const void*)(wsv + k);
        asm volatile("global_load_async_to_lds_b128 %0, %1, off"
                     :
                     : "v"(lds_off), "v"(gaddr)
                     : "memory");
    }
#if __has_builtin(__builtin_amdgcn_s_wait_asynccnt)
    __builtin_amdgcn_s_wait_asynccnt(0);
#else
    asm volatile("s_wait_asynccnt 0" ::: "memory");
#endif
    __syncthreads();

    int lane = threadIdx.x & 31;
    int wv   = threadIdx.x >> 5;
    int tile = blockIdx.x * 8 + wv;          // 4096 tiles total
    int tx   = tile & 63, ty = tile >> 6;    // 64x64 tiles of 4x4 px
    int p    = lane & 15;
    float xx = (float)(tx * 4 + (p & 3));    // absolute pixel coords
    float yy = (float)(ty * 4 + (p >> 2));
    float ctx = (float)(tx * 4) + 1.5f;      // tile centre
    float cty = (float)(ty * 4) + 1.5f;
    float xr = xx - ctx, yr = yy - cty;      // recentred pixel coords
    bool  hi = lane >= 16;

    // B matrix (per-tile constant): K0,K1 in low lanes; K2,K3 in high lanes
    v2f B;
    B[0] = hi ? xr : (xr * xr + yr * yr);
    B[1] = hi ? yr : 1.0f;

    float imr = bg[0], img = bg[1], imb = bg[2];

    for (int c = 0; c < 32; ++c) {
        // A matrix: gaussian m = lane&15 of this chunk
        int   ga = c * 16 + (lane & 15);
        float4 pq = sgs[ga * 3 + 1];         // {n2s, op, u, v}
        float ur = pq.z - ctx, vr = pq.w - cty;
        v2f A;
        A[0] = hi ? (-2.0f * ur) : 1.0f;
        A[1] = hi ? (-2.0f * vr) : (ur * ur + vr * vr);

        v8f cz = {};
        v8f dist = __builtin_amdgcn_wmma_f32_16x16x4_f32(
            false, A, false, B, (short)0, cz, false, false);

        // fold 8-gaussian run into affine map (C, T), in sorted order
        float Cr = 0.0f, Cg = 0.0f, Cb = 0.0f, T = 1.0f;
        int mb = hi ? 8 : 0;
#pragma unroll
        for (int j = 0; j < 8; ++j) {
            int g = c * 16 + mb + j;
            float4 box = sgs[g * 3 + 0];
            float4 par = sgs[g * 3 + 1];
            float4 crgb = sgs[g * 3 + 2];
            bool inb = (xx >= box.x) & (xx < box.y) & (yy >= box.z) & (yy < box.w);
            float e = __expf(dist[j] * par.x);
            float a = fminf(fmaxf(e * par.y, 0.0f), 1.0f);
            a = inb ? a : 0.0f;
            float ia = 1.0f - a;
            Cr = a * crgb.x + ia * Cr;
            Cg = a * crgb.y + ia * Cg;
            Cb = a * crgb.z + ia * Cb;
            T *= ia;
        }

        // merge the two half-wave runs: result = hi ∘ lo
        float oCr = __shfl_xor(Cr, 16, 32);
        float oCg = __shfl_xor(Cg, 16, 32);
        float oCb = __shfl_xor(Cb, 16, 32);
        float oT  = __shfl_xor(T,  16, 32);
        float loCr = hi ? oCr : Cr,  hiCr = hi ? Cr : oCr;
        float loCg = hi ? oCg : Cg,  hiCg = hi ? Cg : oCg;
        float loCb = hi ? oCb : Cb,  hiCb = hi ? Cb : oCb;
        float loT  = hi ? oT  : T,   hiT  = hi ? T  : oT;
        float nCr = hiCr + hiT * loCr;
        float nCg = hiCg + hiT * loCg;
        float nCb = hiCb + hiT * loCb;
        float nT  = hiT * loT;

        imr = nCr + nT * imr;
        img = nCg + nT * img;
        imb = nCb + nT * imb;
    }

    if (!hi) {
        int idx = (ty * 4 + (p >> 2)) * IMG_W + tx * 4 + (p & 3);
        out[idx]                       = imr;
        out[IMG_W * IMG_H + idx]       = img;
        out[2 * IMG_W * IMG_H + idx]   = imb;
    }
}

extern "C" void kernel_launch(void* const* d_in, const int* in_sizes, int n_in,
                              void* d_out, int out_size, void* d_ws, size_t ws_size,
                              hipStream_t stream)
{
    const float* pos = (const float*)d_in[0];
    const float* scl = (const float*)d_in[1];
    // d_in[2] = rotations (unused by reference)
    const float* col = (const float*)d_in[3];
    const float* opa = (const float*)d_in[4];
    const float* vm  = (const float*)d_in[5];
    const float* bg  = (const float*)d_in[6];
    float* ws  = (float*)d_ws;
    float* out = (float*)d_out;

    splat_prep<<<dim3(1), dim3(N_G), 0, stream>>>(pos, scl, col, opa, vm, ws);
    splat_render<<<dim3((IMG_W / 4) * (IMG_H / 4) / 8), dim3(256), 0, stream>>>(ws, bg, out);
}